// RoIAlign_61899068670032
// MI455X (gfx1250) — compile-verified
//
#include <hip/hip_runtime.h>
#include <cstdint>

// Problem constants (from reference)
#define CROP_W 32
#define C_DIM  256
#define W_FD   2048
#define NWAVES 8                   // 256 threads / wave32
#define K_ITERS (C_DIM / NWAVES)   // 32 channel iterations per wave
#define DEPTH   4                  // software-pipeline stages (LDS buffers/wave)

// One block per box. lane = crop position (32 lanes == CROP_W), wave j handles
// channels j, j+8, ..., j+248 -> each wave stores one contiguous 128B line of
// out[m, c, :] per iteration (perfectly coalesced, non-temporal so the 262MB
// output stream does not evict the L2-resident 32MB featuremap).
// The per-lane v_lo/v_hi gathers are staged through LDS with the CDNA5
// async-to-LDS path (ASYNCcnt), 4 stages deep, so each wave keeps up to 8
// gather ops in flight while lerping/storing stage k.
__global__ __launch_bounds__(256)
void roialign1d_async_kernel(const float* __restrict__ fm,
                             const float* __restrict__ boxes,
                             const int*   __restrict__ box_ind,
                             float* __restrict__ out)
{
    __shared__ float stage[NWAVES][DEPTH][2 * CROP_W]; // [wave][buf][0..31=lo | 32..63=hi]

    const int m    = blockIdx.x;
    const int lane = (int)(threadIdx.x & 31u);   // wave32 (gfx1250)
    const int wave = (int)(threadIdx.x >> 5u);

    // Uniform per-block box parameters (scalar float pipeline on gfx1250).
    const float x1 = boxes[2 * m + 0];
    const float x2 = boxes[2 * m + 1];
    const int   b  = box_ind[m];

    // Mirror the reference math exactly.
    const float wdm1    = (float)(W_FD - 1);
    const float spacing = (x2 - x1) / (float)CROP_W;
    const float nx0     = (x1 + spacing * 0.5f - 0.5f) / wdm1;
    const float nw      = spacing * (float)(CROP_W - 1) / wdm1;
    const float in_x    = nx0 * wdm1 + (float)lane * nw * wdm1 / (float)(CROP_W - 1);

    const bool  valid = (in_x >= 0.0f) && (in_x <= wdm1);
    const float lo_f  = floorf(in_x);
    const float frac  = in_x - lo_f;
    const int   lo_i  = (int)fminf(fmaxf(lo_f, 0.0f), wdm1);
    const int   hi_i  = (int)fminf(fmaxf(lo_f + 1.0f, 0.0f), wdm1);

    const float* base  = fm  + (size_t)b * ((size_t)C_DIM * (size_t)W_FD);
    float*       obase = out + (size_t)m * ((size_t)C_DIM * CROP_W) + lane;

    // Low 32 bits of the generic shared-memory address == wave-relative LDS
    // byte offset.
    const uint32_t lds_base = (uint32_t)(uintptr_t)&stage[wave][0][lane];
    const uint32_t buf_sz   = (uint32_t)(2 * CROP_W * sizeof(float));
    const uint32_t hi_off   = (uint32_t)(CROP_W * sizeof(float));

    // Per-lane global byte offsets of the two gather columns within a channel row.
    const uint64_t alo_base = (uint64_t)(uintptr_t)(base + lo_i);
    const uint64_t ahi_base = (uint64_t)(uintptr_t)(base + hi_i);
    const uint64_t row_step = (uint64_t)W_FD * sizeof(float); // +1 channel

    // ---- prologue: fill stages 0..DEPTH-2 (channels wave, wave+8, wave+16)
    #pragma unroll
    for (int k = 0; k < DEPTH - 1; ++k) {
        const uint64_t roff = (uint64_t)(wave + NWAVES * k) * row_step;
        const uint64_t alo  = alo_base + roff;
        const uint64_t ahi  = ahi_base + roff;
        const uint32_t d    = lds_base + (uint32_t)k * buf_sz;
        asm volatile("global_load_async_to_lds_b32 %0, %1, off"
                     :: "v"(d), "v"(alo) : "memory");
        asm volatile("global_load_async_to_lds_b32 %0, %1, off"
                     :: "v"(d + hi_off), "v"(ahi) : "memory");
    }

    // ---- steady state: issue stage k+DEPTH-1, wait for stage k (async loads
    //      complete in order: <=6 outstanding means stage k has landed), lerp,
    //      NT-store one 128B line.
    #pragma unroll 4
    for (int k = 0; k <= K_ITERS - DEPTH; ++k) {
        const int c  = wave + NWAVES * k;
        {
            const uint64_t roff = (uint64_t)(c + NWAVES * (DEPTH - 1)) * row_step;
            const uint64_t alo  = alo_base + roff;
            const uint64_t ahi  = ahi_base + roff;
            const uint32_t d    = lds_base + (uint32_t)((k + DEPTH - 1) & (DEPTH - 1)) * buf_sz;
            asm volatile("global_load_async_to_lds_b32 %0, %1, off"
                         :: "v"(d), "v"(alo) : "memory");
            asm volatile("global_load_async_to_lds_b32 %0, %1, off"
                         :: "v"(d + hi_off), "v"(ahi) : "memory");
        }
        asm volatile("s_wait_asynccnt 6" ::: "memory");

        const int   buf = k & (DEPTH - 1);
        const float vlo = stage[wave][buf][lane];
        const float vhi = stage[wave][buf][lane + CROP_W];
        const float r   = valid ? (vlo * (1.0f - frac) + vhi * frac) : 0.0f;
        __builtin_nontemporal_store(r, obase + (size_t)c * CROP_W);
    }

    // ---- epilogue: drain the last DEPTH-1 stages with waits 4, 2, 0.
    {
        const int k = K_ITERS - 3;
        asm volatile("s_wait_asynccnt 4" ::: "memory");
        const int   buf = k & (DEPTH - 1);
        const float vlo = stage[wave][buf][lane];
        const float vhi = stage[wave][buf][lane + CROP_W];
        const float r   = valid ? (vlo * (1.0f - frac) + vhi * frac) : 0.0f;
        __builtin_nontemporal_store(r, obase + (size_t)(wave + NWAVES * k) * CROP_W);
    }
    {
        const int k = K_ITERS - 2;
        asm volatile("s_wait_asynccnt 2" ::: "memory");
        const int   buf = k & (DEPTH - 1);
        const float vlo = stage[wave][buf][lane];
        const float vhi = stage[wave][buf][lane + CROP_W];
        const float r   = valid ? (vlo * (1.0f - frac) + vhi * frac) : 0.0f;
        __builtin_nontemporal_store(r, obase + (size_t)(wave + NWAVES * k) * CROP_W);
    }
    {
        const int k = K_ITERS - 1;
        asm volatile("s_wait_asynccnt 0" ::: "memory");
        const int   buf = k & (DEPTH - 1);
        const float vlo = stage[wave][buf][lane];
        const float vhi = stage[wave][buf][lane + CROP_W];
        const float r   = valid ? (vlo * (1.0f - frac) + vhi * frac) : 0.0f;
        __builtin_nontemporal_store(r, obase + (size_t)(wave + NWAVES * k) * CROP_W);
    }
}

extern "C" void kernel_launch(void* const* d_in, const int* in_sizes, int n_in,
                              void* d_out, int out_size, void* d_ws, size_t ws_size,
                              hipStream_t stream) {
    const float* fm      = (const float*)d_in[0];   // (16, 256, 2048) f32
    const float* boxes   = (const float*)d_in[1];   // (8000, 2) f32
    const int*   box_ind = (const int*)d_in[2];     // (8000,) i32
    float*       out     = (float*)d_out;           // (8000, 256, 32) f32

    const int m_boxes = in_sizes[2];                // 8000
    (void)n_in; (void)out_size; (void)d_ws; (void)ws_size;

    roialign1d_async_kernel<<<m_boxes, 256, 0, stream>>>(fm, boxes, box_ind, out);
}